// GCNEncoder_86328842650113
// MI455X (gfx1250) — compile-verified
//
#include <hip/hip_runtime.h>

// ---------------------------------------------------------------------------
// 2-layer GCN for MI455X (gfx1250, wave32).
// Roofline: edge aggregation moves ~2.5 GB (gather + fp32 atomics), all
// L2-resident (h/a buffers = 51.2/25.6 MB << 192 MB L2) -> ~110us floor.
// GEMMs (5 GFLOP total) are noise; run them on the fp32 WMMA pipe
// (V_WMMA_F32_16X16X4_F32) with fully-unrolled K (template constants) so
// every load is base+immediate, A staged in LDS (conflict-free padded
// stride), B from a pre-transposed weight copy as single b64 loads.
// ---------------------------------------------------------------------------

typedef __attribute__((ext_vector_type(2))) float v2f;
typedef __attribute__((ext_vector_type(8))) float v8f;

// ---- out-degree: deg[src[e]] += 1 ----------------------------------------
__global__ void k_degree(const int* __restrict__ src, float* __restrict__ deg, int E) {
    int e = blockIdx.x * blockDim.x + threadIdx.x;
    if (e < E) unsafeAtomicAdd(&deg[src[e]], 1.0f);
}

// ---- deg -> 1/max(deg,1) in place ----------------------------------------
__global__ void k_deg_inv(float* __restrict__ deg, int n) {
    int i = blockIdx.x * blockDim.x + threadIdx.x;
    if (i < n) deg[i] = 1.0f / fmaxf(deg[i], 1.0f);
}

// ---- Wt[c][k] = W[k][c]  (tiny one-shot transpose, K*C <= 16K elems) ------
template <int K, int C>
__global__ void k_transpose(const float* __restrict__ W, float* __restrict__ Wt) {
    int i = blockIdx.x * blockDim.x + threadIdx.x;
    if (i < K * C) {
        int k = i / C, c = i % C;          // C is a power-of-2 constant
        Wt[c * K + k] = W[i];
    }
}

// ---- H[r][c] = deg_inv[r] * sum_k X[r][k] * W[k][c]  (fp32 WMMA) ----------
// Block = 16 rows; wave w computes the 16x16 tile at cols [16w,16w+16).
// A tile (16 x K fp32, 8KB) staged in LDS with row stride K+4 floats:
//   - co-op float4 global loads/LDS stores stay 16B aligned (528 = 33*16)
//   - per-lane ds_load_b64 at stride 528B -> banks (4m + ka) % 64, distinct.
// A-frag (16x4 f32, ISA 7.12.2): lane L: m = L%16, lane-half adds K+2;
// C/D frag: VGPR v holds M = v + 8*(L/16), N = L%16.
template <int K, int C>
__global__ void k_gemm_scale(const float* __restrict__ X, const float* __restrict__ Wt,
                             const float* __restrict__ deg_inv, float* __restrict__ H,
                             int nrows) {
    constexpr int LDA = K + 4;                 // padded LDS row stride (floats)
    __shared__ float sA[16 * LDA];

    const int tid      = threadIdx.x;
    const int nthreads = 32 * (C / 16);
    const int rowBase  = blockIdx.x * 16;

    // cooperative load: 16 rows x K cols of X -> LDS (float4 granularity)
    constexpr int NV4 = 16 * K / 4;
    for (int i = tid; i < NV4; i += nthreads) {
        const int rr  = i / (K / 4);
        const int kk4 = i % (K / 4);
        int r = rowBase + rr;
        if (r >= nrows) r = nrows - 1;         // clamp; N%16==0 so usually no-op
        const float4 v = *(const float4*)(X + (size_t)r * K + 4 * kk4);
        *(float4*)(&sA[rr * LDA + 4 * kk4]) = v;
    }
    __syncthreads();                            // uniform: EXEC stays all-ones

    const int lane    = tid & 31;
    const int wv      = tid >> 5;
    const int colBase = wv * 16;
    const int m       = lane & 15;
    const int lh      = lane >> 4;              // lane-half -> K sub-pair (+2)

    const float* sa  = &sA[m * LDA + 2 * lh];
    const float* wtc = Wt + (size_t)(colBase + m) * K + 2 * lh;

    v8f c = {0.f, 0.f, 0.f, 0.f, 0.f, 0.f, 0.f, 0.f};
#pragma unroll
    for (int k = 0; k < K; k += 4) {
        const v2f a = *(const v2f*)(sa + k);    // ds_load_b64, immediate offset
        const v2f b = *(const v2f*)(wtc + k);   // global_load_b64, immediate offset
        // (neg_a, A, neg_b, B, c_mod, C, reuse_a, reuse_b)
        c = __builtin_amdgcn_wmma_f32_16x16x4_f32(false, a, false, b,
                                                  (short)0, c, false, false);
    }

#pragma unroll
    for (int v = 0; v < 8; ++v) {
        const int row = rowBase + v + 8 * lh;
        if (row < nrows)
            H[(size_t)row * C + colBase + m] = c[v] * deg_inv[row];
    }
}

// ---- out[dst[e]] += H[src[e]]   (one wave per edge) -----------------------
// C=128: 4 floats/lane (b128 gather); C=64: 2 floats/lane (b64 gather).
template <int C>
__global__ void k_scatter_add(const float* __restrict__ H, const int* __restrict__ src,
                              const int* __restrict__ dst, float* __restrict__ out, int E) {
    constexpr int PL = C / 32;                  // floats per lane
    const int lane = threadIdx.x & 31;
    const int e = blockIdx.x * (blockDim.x >> 5) + (threadIdx.x >> 5);
    if (e >= E) return;
    const int s = src[e];
    const int d = dst[e];
    const float* hs = H + (size_t)s * C + lane * PL;
    float*       od = out + (size_t)d * C + lane * PL;
    float v[PL];
#pragma unroll
    for (int j = 0; j < PL; ++j) v[j] = hs[j];               // coalesced vector load
#pragma unroll
    for (int j = 0; j < PL; ++j) unsafeAtomicAdd(&od[j], v[j]); // global_atomic_add_f32
}

// ---- A[i] = (A[i] + bias[i % C]) [, relu] ---------------------------------
template <bool RELU, int C>
__global__ void k_bias(float* __restrict__ A, const float* __restrict__ bias, int n) {
    int i = blockIdx.x * blockDim.x + threadIdx.x;
    if (i >= n) return;
    float v = A[i] + bias[i & (C - 1)];
    if (RELU) v = fmaxf(v, 0.f);
    A[i] = v;
}

extern "C" void kernel_launch(void* const* d_in, const int* in_sizes, int n_in,
                              void* d_out, int out_size, void* d_ws, size_t ws_size,
                              hipStream_t stream) {
    const float* x  = (const float*)d_in[0];
    const int*   ei = (const int*)d_in[1];
    const float* W1 = (const float*)d_in[2];
    const float* b1 = (const float*)d_in[3];
    const float* W2 = (const float*)d_in[4];
    const float* b2 = (const float*)d_in[5];
    float* out = (float*)d_out;

    constexpr int IN_C = 128, HID_C = 128, OUT_C = 64;
    const int N = in_sizes[0] / IN_C;
    const int E = in_sizes[1] / 2;
    const int* src = ei;
    const int* dst = ei + E;

    // workspace: deg | h1 | a1 | h2 | Wt1 | Wt2   (~128.5 MB)
    auto a256 = [](size_t v) { return (v + 255) & ~(size_t)255; };
    char* ws = (char*)d_ws;
    size_t off = 0;
    float* deg = (float*)(ws + off); off += a256((size_t)N * 4);
    float* h1  = (float*)(ws + off); off += a256((size_t)N * HID_C * 4);
    float* a1  = (float*)(ws + off); off += a256((size_t)N * HID_C * 4);
    float* h2  = (float*)(ws + off); off += a256((size_t)N * OUT_C * 4);
    float* Wt1 = (float*)(ws + off); off += a256((size_t)IN_C * HID_C * 4);
    float* Wt2 = (float*)(ws + off); off += a256((size_t)HID_C * OUT_C * 4);
    (void)off; (void)ws_size; (void)n_in; (void)out_size;

    // zero accumulation targets (harness poisons buffers with 0xAA)
    hipMemsetAsync(deg, 0, (size_t)N * 4, stream);
    hipMemsetAsync(a1,  0, (size_t)N * HID_C * 4, stream);
    hipMemsetAsync(out, 0, (size_t)N * OUT_C * 4, stream);

    // weight transposes (so B fragments are contiguous b64 loads)
    k_transpose<IN_C, HID_C><<<(IN_C * HID_C + 255) / 256, 256, 0, stream>>>(W1, Wt1);
    k_transpose<HID_C, OUT_C><<<(HID_C * OUT_C + 255) / 256, 256, 0, stream>>>(W2, Wt2);

    // degrees
    k_degree <<<(E + 255) / 256, 256, 0, stream>>>(src, deg, E);
    k_deg_inv<<<(N + 255) / 256, 256, 0, stream>>>(deg, N);

    // layer 1: h1 = (x @ W1) * deg_inv ; a1 = relu(scatter_add(h1) + b1)
    k_gemm_scale<IN_C, HID_C><<<(N + 15) / 16, 32 * (HID_C / 16), 0, stream>>>(x, Wt1, deg, h1, N);
    k_scatter_add<HID_C><<<(E + 7) / 8, 256, 0, stream>>>(h1, src, dst, a1, E);
    k_bias<true, HID_C><<<((size_t)N * HID_C + 255) / 256, 256, 0, stream>>>(a1, b1, N * HID_C);

    // layer 2: h2 = (a1 @ W2) * deg_inv ; out = scatter_add(h2) + b2
    k_gemm_scale<HID_C, OUT_C><<<(N + 15) / 16, 32 * (OUT_C / 16), 0, stream>>>(a1, Wt2, deg, h2, N);
    k_scatter_add<OUT_C><<<(E + 7) / 8, 256, 0, stream>>>(h2, src, dst, out, E);
    k_bias<false, OUT_C><<<((size_t)N * OUT_C + 255) / 256, 256, 0, stream>>>(out, b2, N * OUT_C);
}